// AttnReweight_27307402068263
// MI455X (gfx1250) — compile-verified
//
#include <hip/hip_runtime.h>
#include <stdint.h>

// Problem constants (match reference).
constexpr int Bc  = 2;
constexpr int HDc = 4;
constexpr int Hc  = 128;
constexpr int Wc  = 128;
constexpr int KSc = 7;
constexpr int Kc  = 49;
constexpr int NSP = 9;
constexpr int Sc  = 64;

// Tiling: 8x8 pixels per block, 4 heads -> 256 threads.
constexpr int TH   = 8;
constexpr int TW   = 8;
constexpr int HALO = 3;
constexpr int LH   = TH + 2 * HALO;          // 14
constexpr int LW   = TW + 2 * HALO;          // 14
constexpr int CHS  = LH * LW + 1;            // 197 (odd pad: decorrelate LDS banks)
constexpr int STAGE_ITERS = (Sc * LH * LW) / 256;   // 12544/256 = 49 exactly

__global__ __launch_bounds__(256)
void attn_reweight_kernel(const float* __restrict__ attn,
                          const float* __restrict__ sims,
                          const int*   __restrict__ sinds,
                          float*       __restrict__ out)
{
    __shared__ float tile[Sc * CHS];         // 50,432 B

    const int b  = blockIdx.z;
    const int h0 = blockIdx.y * TH;
    const int w0 = blockIdx.x * TW;
    const int hb = h0 - HALO;                // tile halo origin (may be negative)
    const int wb = w0 - HALO;

    const unsigned long long simsBase =
        (unsigned long long)(uintptr_t)(sims + (size_t)b * Sc * Hc * Wc);

    // ------------------------------------------------------------------
    // Stage sims[b, 0:64, hb:hb+14, wb:wb+14] (edge-clamped; clamped slots
    // are never read by the compute phase) into LDS via CDNA5 async
    // global->LDS DMA. Constant trip count (49), partially unrolled so
    // several async transfers are in flight per loop iteration.
    // ------------------------------------------------------------------
    const int tid = threadIdx.x;
    #pragma unroll 4
    for (int it = 0; it < STAGE_ITERS; ++it) {
        int t   = tid + it * 256;
        int s   = t / (LH * LW);
        int rem = t - s * (LH * LW);
        int u   = rem / LW;
        int v   = rem - u * LW;
        int ch  = hb + u; ch = ch < 0 ? 0 : (ch > Hc - 1 ? Hc - 1 : ch);
        int cw  = wb + v; cw = cw < 0 ? 0 : (cw > Wc - 1 ? Wc - 1 : cw);
        unsigned goff = (unsigned)((s * Hc * Wc + ch * Wc + cw) * 4);
        unsigned ldsb = (unsigned)(size_t)&tile[s * CHS + u * LW + v];
        asm volatile("global_load_async_to_lds_b32 %0, %1, %2"
                     :: "v"(ldsb), "v"(goff), "s"(simsBase) : "memory");
    }
    // Each wave drains its own async transfers, then the workgroup syncs
    // so every wave sees the fully staged tile.
    asm volatile("s_wait_asynccnt 0x0" ::: "memory");
    __syncthreads();

    // ------------------------------------------------------------------
    // Thread mapping: 4 consecutive lanes = 4 heads of one pixel, so their
    // LDS gathers of p hit identical addresses (LDS broadcast).
    // ------------------------------------------------------------------
    const int hd  = tid & 3;
    const int pix = tid >> 2;                // 0..63
    const int wl  = pix & (TW - 1);          // 0..7
    const int hl  = pix >> 3;                // 0..7
    const int h   = h0 + hl;
    const int w   = w0 + wl;

    // Clipped 7x7 window start (matches reference _neighbor_coords).
    int hs = h - HALO; hs = hs < 0 ? 0 : (hs > Hc - KSc ? Hc - KSc : hs);
    int ws = w - HALO; ws = ws < 0 ? 0 : (ws > Wc - KSc ? Wc - KSc : ws);
    const int lbase = (hs - hb) * LW + (ws - wb);            // window origin in tile
    const int lcent = (hl + HALO) * LW + (wl + HALO);        // pixel center in tile

    // Superpixel channels + center weights pi[n].
    const int* sp = sinds + (((size_t)b * Hc + h) * Wc + w) * NSP;
    int   sb[NSP];
    float pi[NSP];
    #pragma unroll
    for (int n = 0; n < NSP; ++n) {
        int s = sp[n];
        sb[n] = s * CHS;
        pi[n] = tile[sb[n] + lcent];
    }

    // Load attention row, compute a = exp(attn - max).
    const float* arow =
        attn + ((((size_t)b * HDc + hd) * Hc + h) * Wc + w) * (size_t)Kc;
    float a[Kc];
    float c = -3.0e38f;
    #pragma unroll
    for (int k = 0; k < Kc; ++k) { a[k] = arow[k]; c = fmaxf(c, a[k]); }
    #pragma unroll
    for (int k = 0; k < Kc; ++k) a[k] = __expf(a[k] - c);

    // Pass 1: denom[n] = sum_k a[k] * p[n,k]  ->  coef[n] = pi[n]/(eps+denom).
    // Row-contiguous LDS reads (mergeable into ds_load_2addr/b64).
    float coef[NSP];
    #pragma unroll
    for (int n = 0; n < NSP; ++n) {
        const float* tp = tile + sb[n] + lbase;
        float dn = 0.0f;
        #pragma unroll
        for (int k = 0; k < Kc; ++k)
            dn += a[k] * tp[(k / KSc) * LW + (k % KSc)];
        coef[n] = pi[n] / (1e-10f + dn);
    }

    // Pass 2: acc[k] = sum_n coef[n] * p[n,k], n-outer so the inner k reads
    // are row-contiguous per channel (vectorizable LDS access like pass 1).
    float acc[Kc];
    #pragma unroll
    for (int k = 0; k < Kc; ++k) acc[k] = 0.0f;
    #pragma unroll
    for (int n = 0; n < NSP; ++n) {
        const float cn = coef[n];
        const float* tp = tile + sb[n] + lbase;
        #pragma unroll
        for (int k = 0; k < Kc; ++k)
            acc[k] += cn * tp[(k / KSc) * LW + (k % KSc)];
    }

    // out[k] = a[k] * acc[k], streamed out (compiler merges to wide stores).
    float* orow =
        out + ((((size_t)b * HDc + hd) * Hc + h) * Wc + w) * (size_t)Kc;
    #pragma unroll
    for (int k = 0; k < Kc; ++k) orow[k] = a[k] * acc[k];
}

extern "C" void kernel_launch(void* const* d_in, const int* in_sizes, int n_in,
                              void* d_out, int out_size, void* d_ws, size_t ws_size,
                              hipStream_t stream) {
    const float* attn  = (const float*)d_in[0];
    const float* sims  = (const float*)d_in[1];
    const int*   sinds = (const int*)d_in[2];
    float* out = (float*)d_out;

    dim3 grid(Wc / TW, Hc / TH, Bc);   // (16, 16, 2)
    attn_reweight_kernel<<<grid, 256, 0, stream>>>(attn, sims, sinds, out);
}